// WindowedLinear_85504208929310
// MI455X (gfx1250) — compile-verified
//
#include <hip/hip_runtime.h>

// ---------------- problem constants (from reference) ----------------
#define BSZ      1024
#define D1       50
#define D2       32
#define FEATDIM  20000
#define OUTDIM   2
#define NWIN     4

// ---------------- tiling ----------------
#define ROWS_PER_WG     16            // M tile (batch rows per workgroup)
#define FEAT_CHUNK      5000          // features per (window, chunk) tile
#define CHUNKS_PER_WIN  4             // 20000 / 5000
#define KPAD            5120          // FEAT_CHUNK padded to multiple of 256
#define KWORDS          (KPAD / 32)   // 160 bitmask dwords per row
#define NTHREADS        256           // 8 waves of 32
#define NWAVES          (NTHREADS / 32)
#define GROUPS_PER_WAVE (KWORDS / NWAVES)   // 20 (compile-time constant trip count)

typedef __attribute__((ext_vector_type(8))) float v8f;
typedef __attribute__((ext_vector_type(2))) float v2f;

// Seed output with bias (harness poisons d_out; we accumulate atomically after).
__global__ void wl_init_out(float* __restrict__ out, const float* __restrict__ bias, int n) {
    int i = blockIdx.x * blockDim.x + threadIdx.x;
    if (i < n) out[i] = bias[i & 1];
}

__global__ __launch_bounds__(NTHREADS)
void wl_wmma_kernel(const int*   __restrict__ concept_t,   // (BSZ, D1, D2) int32
                    const int*   __restrict__ times,       // (BSZ, D1) int32
                    const int*   __restrict__ cutoffs,     // (NWIN,) int32
                    const float* __restrict__ weight,      // (OUTDIM, NWIN*FEATDIM) f32
                    float*       __restrict__ out)         // (BSZ, OUTDIM) f32
{
    __shared__ unsigned int bits[ROWS_PER_WG][KWORDS];   // 10 KB feature bitsets
    __shared__ __align__(16) float wlds[OUTDIM][KPAD];   // 40 KB weight slice
    __shared__ __align__(16) float zpad[32];             // zero B source for lanes N>=2
    __shared__ int          tlds[ROWS_PER_WG * D1];      // 3.2 KB times
    __shared__ float        outacc[ROWS_PER_WG][OUTDIM];

    const int tid     = threadIdx.x;
    const int rowBase = blockIdx.x * ROWS_PER_WG;
    const int chunk   = blockIdx.y;                      // 0..15
    const int w       = chunk / CHUNKS_PER_WIN;          // window id
    const int fbase   = (chunk % CHUNKS_PER_WIN) * FEAT_CHUNK;
    const int cutoff  = cutoffs[w];

    // ---- phase 1: init LDS ----
    for (int i = tid; i < ROWS_PER_WG * KWORDS; i += NTHREADS)
        (&bits[0][0])[i] = 0u;
    for (int i = tid; i < OUTDIM * KPAD; i += NTHREADS) {
        int o = i / KPAD, k = i - o * KPAD;
        float v = 0.0f;
        if (k < FEAT_CHUNK)
            v = weight[o * (NWIN * FEATDIM) + w * FEATDIM + fbase + k];
        (&wlds[0][0])[i] = v;
    }
    for (int i = tid; i < ROWS_PER_WG * D1; i += NTHREADS)
        tlds[i] = times[rowBase * D1 + i];
    if (tid < 32) zpad[tid] = 0.0f;
    if (tid < ROWS_PER_WG * OUTDIM)
        (&outacc[0][0])[tid] = 0.0f;
    __syncthreads();

    // ---- phase 2: scatter concept codes into bitsets (dedup via OR) ----
    const int total = ROWS_PER_WG * D1 * D2;             // 25600
    for (int e = tid; e < total; e += NTHREADS) {
        int r   = e / (D1 * D2);
        int rem = e - r * (D1 * D2);
        int vi  = rem >> 5;                              // visit index (rem / D2)
        int t   = tlds[r * D1 + vi];
        int cc  = concept_t[(rowBase + r) * (D1 * D2) + rem];
        bool keep = (t >= cutoff) && (t >= 0);
        int f  = keep ? cc : 0;                          // masked entries collapse to 0
        int lf = f - fbase;
        if ((unsigned)lf < (unsigned)FEAT_CHUNK)
            atomicOr(&bits[r][lf >> 5], 1u << (lf & 31));
    }
    __syncthreads();

    // ---- phase 3: WMMA contraction ----
    // Each wave owns GROUPS_PER_WAVE groups of 32 K-values; per group:
    //   1x ds_load_b32 (bitset) + 8x ds_load_b64 (weight pairs), then 8 WMMAs.
    const int lane = tid & 31;
    const int wave = tid >> 5;
    const int lh   = lane >> 4;     // lane half: picks K pair {0,1} vs {2,3} (f32 A/B layout)
    const int ml   = lane & 15;     // A: row M;  B: col N
    const int lh2  = lh * 2;
    v8f c0 = {}, c1 = {};

#pragma unroll 2
    for (int g = 0; g < GROUPS_PER_WAVE; ++g) {
        const int gg    = g * NWAVES + wave;             // group index 0..159
        const int kbase = gg * 32;
        const unsigned int bw = bits[ml][gg];            // 32 feature bits for row ml
        // B source: real weight row for lanes N<2, zero pad otherwise (no per-step masks)
        const v2f* wp = (ml < OUTDIM) ? (const v2f*)&wlds[ml][kbase + lh2]
                                      : (const v2f*)zpad;
        v2f bvec[8];
#pragma unroll
        for (int i = 0; i < 8; ++i)
            bvec[i] = wp[i * 2];                         // ds_load_b64, imm offsets i*16B
#pragma unroll
        for (int i = 0; i < 8; ++i) {
            const int bp = i * 4 + lh2;
            v2f a;
            a.x = (float)((bw >> bp) & 1u);
            a.y = (float)((bw >> (bp + 1)) & 1u);
            if (i & 1)
                c1 = __builtin_amdgcn_wmma_f32_16x16x4_f32(false, a, false, bvec[i],
                                                           (short)0, c1, false, false);
            else
                c0 = __builtin_amdgcn_wmma_f32_16x16x4_f32(false, a, false, bvec[i],
                                                           (short)0, c0, false, false);
        }
    }
    v8f c = c0 + c1;

    // ---- phase 4: reduce the 8 wave-partials inside the workgroup ----
    // C layout: VGPR v -> M = v (lanes 0-15) or 8+v (lanes 16-31), N = ml
    if (ml < OUTDIM) {
#pragma unroll
        for (int v = 0; v < 8; ++v) {
            int m = lh ? (8 + v) : v;
            atomicAdd(&outacc[m][ml], c[v]);
        }
    }
    __syncthreads();

    // ---- phase 5: accumulate chunk partial into global output ----
    if (tid < ROWS_PER_WG * OUTDIM) {
        int r = tid >> 1, o = tid & 1;
        atomicAdd(&out[(rowBase + r) * OUTDIM + o], outacc[r][o]);
    }
}

extern "C" void kernel_launch(void* const* d_in, const int* in_sizes, int n_in,
                              void* d_out, int out_size, void* d_ws, size_t ws_size,
                              hipStream_t stream) {
    const int*   concept_t = (const int*)  d_in[0];   // (1024,50,32)
    const int*   times     = (const int*)  d_in[1];   // (1024,50)
    const int*   cutoffs   = (const int*)  d_in[2];   // (4,)
    const float* weight    = (const float*)d_in[3];   // (2, 80000)
    const float* bias      = (const float*)d_in[4];   // (2,)
    float*       out       = (float*)      d_out;     // (1024, 2)

    const int n = BSZ * OUTDIM;                        // 2048
    wl_init_out<<<(n + 255) / 256, 256, 0, stream>>>(out, bias, n);

    dim3 grid(BSZ / ROWS_PER_WG, NWIN * CHUNKS_PER_WIN);   // 64 x 16 = 1024 workgroups
    wl_wmma_kernel<<<grid, NTHREADS, 0, stream>>>(concept_t, times, cutoffs, weight, out);
}